// SAGENet_81312320848105
// MI455X (gfx1250) — compile-verified
//
#include <hip/hip_runtime.h>
#include <hip/hip_bf16.h>

typedef __attribute__((ext_vector_type(2))) float v2f;
typedef __attribute__((ext_vector_type(8))) float v8f;

#define NN 100000
#define NE 1600000
#define NF 128
#define NH 16
#define NC 64

// ---- degree: deg[d] += 1 per edge -------------------------------------------
__global__ void deg_kernel(const int* __restrict__ dst, float* __restrict__ deg) {
  int e = blockIdx.x * blockDim.x + threadIdx.x;
  if (e < NE) atomicAdd(&deg[dst[e]], 1.0f);
}

// ---- layer1 transform: t1 = x@w1_l, r1 = x@w1_r  (WMMA f32 16x16x4) ---------
// One wave32 per 16-row tile. A 16x4 layout (ISA 7.12.2): lane<16 -> K=k,k+1;
// lane>=16 -> K=k+2,k+3 (M = lane&15). B 4x16: same K split, N = lane&15.
// C/D 16x16: lane<16 -> M=r (vgpr r), lane>=16 -> M=8+r, N = lane&15.
__global__ __launch_bounds__(32)
void lin1_kernel(const float* __restrict__ x, const float* __restrict__ w_l,
                 const float* __restrict__ w_r,
                 float* __restrict__ t1, float* __restrict__ r1) {
  const int m0   = blockIdx.x * 16;
  const int lane = threadIdx.x;
  const int n16  = lane & 15;
  const int koff = (lane >> 4) * 2;
  const int row  = m0 + n16;
  const float* xr = x + (size_t)row * NF;
  v8f cl = {}; v8f cr = {};
  for (int k = 0; k < NF; k += 4) {
    const int kk = k + koff;
    v2f a;  a.x  = xr[kk];                 a.y  = xr[kk + 1];
    v2f bl; bl.x = w_l[kk * NH + n16];     bl.y = w_l[(kk + 1) * NH + n16];
    v2f br; br.x = w_r[kk * NH + n16];     br.y = w_r[(kk + 1) * NH + n16];
    cl = __builtin_amdgcn_wmma_f32_16x16x4_f32(false, a, false, bl, (short)0, cl, false, false);
    cr = __builtin_amdgcn_wmma_f32_16x16x4_f32(false, a, false, br, (short)0, cr, false, false);
  }
  const int mbase = (lane >> 4) * 8;
  for (int r = 0; r < 8; ++r) {
    t1[(size_t)(m0 + mbase + r) * NH + n16] = cl[r];
    r1[(size_t)(m0 + mbase + r) * NH + n16] = cr[r];
  }
}

// ---- scatter-add 16-wide features along edges (4 floats / thread) -----------
__global__ void agg_kernel(const int* __restrict__ src, const int* __restrict__ dst,
                           const float* __restrict__ feat, float* __restrict__ agg) {
  long long t = (long long)blockIdx.x * blockDim.x + threadIdx.x;
  if (t >= (long long)NE * 4) return;
  const int e = (int)(t >> 2);
  const int q = (int)(t & 3);
  const int s = src[e], d = dst[e];
  const float4 v = ((const float4*)feat)[(size_t)s * 4 + q];
  float* p = agg + (size_t)d * NH + q * 4;
  atomicAdd(p + 0, v.x);
  atomicAdd(p + 1, v.y);
  atomicAdd(p + 2, v.z);
  atomicAdd(p + 3, v.w);
}

// ---- h = relu(agg1/deg + b1 + r1), in place over agg1 -----------------------
__global__ void combine1_kernel(const float* __restrict__ deg, const float* __restrict__ r1,
                                const float* __restrict__ b1, float* __restrict__ agg_h) {
  int t = blockIdx.x * blockDim.x + threadIdx.x;
  if (t >= NN * NH) return;
  const int i = t >> 4, f = t & 15;
  const float dinv = 1.0f / fmaxf(deg[i], 1.0f);
  const float v = agg_h[t] * dinv + b1[f] + r1[t];
  agg_h[t] = fmaxf(v, 0.0f);
}

// ---- layer2: out = [agg2/deg | h] (Nx32) @ [w2_l;w2_r] (32x64) + b2 ---------
__global__ __launch_bounds__(32)
void lin2_kernel(const float* __restrict__ agg2, const float* __restrict__ h,
                 const float* __restrict__ deg,
                 const float* __restrict__ w_l, const float* __restrict__ w_r,
                 const float* __restrict__ b2, float* __restrict__ out) {
  const int m0   = blockIdx.x * 16;
  const int lane = threadIdx.x;
  const int n16  = lane & 15;
  const int koff = (lane >> 4) * 2;
  const int row  = m0 + n16;
  const float dinv = 1.0f / fmaxf(deg[row], 1.0f);
  v8f acc[4] = {};
  for (int k = 0; k < 32; k += 4) {
    const int kk = k + koff;               // kk, kk+1 stay in the same half
    v2f a;
    const float* w;
    if (kk < 16) {
      a.x = agg2[(size_t)row * NH + kk] * dinv;
      a.y = agg2[(size_t)row * NH + kk + 1] * dinv;
      w = w_l + (size_t)kk * NC;
    } else {
      a.x = h[(size_t)row * NH + (kk - 16)];
      a.y = h[(size_t)row * NH + (kk - 15)];
      w = w_r + (size_t)(kk - 16) * NC;
    }
    #pragma unroll
    for (int nt = 0; nt < 4; ++nt) {
      v2f b; b.x = w[nt * 16 + n16]; b.y = w[NC + nt * 16 + n16];
      acc[nt] = __builtin_amdgcn_wmma_f32_16x16x4_f32(false, a, false, b, (short)0, acc[nt], false, false);
    }
  }
  const int mbase = (lane >> 4) * 8;
  #pragma unroll
  for (int nt = 0; nt < 4; ++nt) {
    const float bias = b2[nt * 16 + n16];
    #pragma unroll
    for (int r = 0; r < 8; ++r)
      out[(size_t)(m0 + mbase + r) * NC + nt * 16 + n16] = acc[nt][r] + bias;
  }
}

// ---- log_softmax over rows of 64, one wave32 per row, in place --------------
__global__ __launch_bounds__(256)
void lsm_kernel(float* __restrict__ out) {
  const int wave = (blockIdx.x * blockDim.x + threadIdx.x) >> 5;
  const int lane = threadIdx.x & 31;
  if (wave >= NN) return;
  float* row = out + (size_t)wave * NC;
  const float a = row[lane];
  const float b = row[lane + 32];
  float m = fmaxf(a, b);
  for (int off = 16; off > 0; off >>= 1) m = fmaxf(m, __shfl_xor(m, off, 32));
  float s = expf(a - m) + expf(b - m);
  for (int off = 16; off > 0; off >>= 1) s += __shfl_xor(s, off, 32);
  const float lse = m + logf(s);
  row[lane]      = a - lse;
  row[lane + 32] = b - lse;
}

extern "C" void kernel_launch(void* const* d_in, const int* in_sizes, int n_in,
                              void* d_out, int out_size, void* d_ws, size_t ws_size,
                              hipStream_t stream) {
  const float* x    = (const float*)d_in[0];
  const int*   ei   = (const int*)d_in[1];
  const int*   src  = ei;
  const int*   dst  = ei + NE;
  const float* w1_l = (const float*)d_in[2];
  const float* b1   = (const float*)d_in[3];
  const float* w1_r = (const float*)d_in[4];
  const float* w2_l = (const float*)d_in[5];
  const float* b2   = (const float*)d_in[6];
  const float* w2_r = (const float*)d_in[7];
  float* out = (float*)d_out;

  float* deg  = (float*)d_ws;                    // N
  float* agg1 = deg  + NN;                       // N*16 (becomes h after combine1)
  float* agg2 = agg1 + (size_t)NN * NH;          // N*16
  float* t1   = agg2 + (size_t)NN * NH;          // N*16
  float* r1   = t1   + (size_t)NN * NH;          // N*16

  // zero deg + agg1 + agg2 (accumulators); t1/r1 are fully overwritten
  hipMemsetAsync(deg, 0, (size_t)NN * (1 + 2 * NH) * sizeof(float), stream);

  deg_kernel<<<(NE + 255) / 256, 256, 0, stream>>>(dst, deg);
  lin1_kernel<<<NN / 16, 32, 0, stream>>>(x, w1_l, w1_r, t1, r1);
  agg_kernel<<<(NE * 4 + 255) / 256, 256, 0, stream>>>(src, dst, t1, agg1);
  combine1_kernel<<<(NN * NH + 255) / 256, 256, 0, stream>>>(deg, r1, b1, agg1);
  agg_kernel<<<(NE * 4 + 255) / 256, 256, 0, stream>>>(src, dst, agg1, agg2);
  lin2_kernel<<<NN / 16, 32, 0, stream>>>(agg2, agg1, deg, w2_l, w2_r, b2, out);
  lsm_kernel<<<(NN * 32 + 255) / 256, 256, 0, stream>>>(out);
}